// EdgeModel_74663711473943
// MI455X (gfx1250) — compile-verified
//
#include <hip/hip_runtime.h>

typedef __attribute__((ext_vector_type(2))) float f2;
typedef __attribute__((ext_vector_type(8))) float v8f;

#define WAVES_PER_BLOCK 8
#define TILE 16
#define TILES_PER_WAVE 8
#define LDS1_STRIDE 36   // K padded 35->36; 36*M mod 64 distinct for M<16 (gcd(9,16)=1)
#define LDS2_STRIDE 18   // 16 -> 18; 18*M mod 64 distinct for M<16 (gcd(9,32)=1); keeps 8B align

// LDS column layout (permuted concat, pad at k=15):
//   [0..9]  x_s[src]   [10..14] x_t[tgt]   [15] zero-pad
//   [16..25] edge_attr [26..35] u[batch_e]
// W1 row for LDS column k:  k<15 -> k ; k==15 -> (zero) ; k>15 -> k-1
__global__ __launch_bounds__(256) void edge_mlp_wmma(
    const float* __restrict__ x_s, const float* __restrict__ x_t,
    const int* __restrict__ edge_index, const float* __restrict__ edge_attr,
    const float* __restrict__ u, const int* __restrict__ batch_e,
    const float* __restrict__ W1, const float* __restrict__ b1,
    const float* __restrict__ W2, const float* __restrict__ b2,
    float* __restrict__ out, int E, int tiles)
{
    __shared__ float lds1[WAVES_PER_BLOCK][TILE * LDS1_STRIDE];
    __shared__ float lds2[WAVES_PER_BLOCK][TILE * LDS2_STRIDE];

    const int lane = threadIdx.x & 31;
    const int wave = threadIdx.x >> 5;
    float* t1 = lds1[wave];
    float* t2 = lds2[wave];

    const int total_waves = gridDim.x * WAVES_PER_BLOCK;          // uniform
    const int iters = (tiles + total_waves - 1) / total_waves;    // uniform
    const int wave_id = blockIdx.x * WAVES_PER_BLOCK + wave;

    const int n     = lane & 15;          // B/C column held by this lane
    const int m     = lane & 15;          // A row held by this lane
    const int khalf = (lane >> 4) * 2;    // lanes 16-31 hold K+2,K+3

    // ---- hoist W1/W2/b1/b2 into B-fragment registers (once per wave) ----
    f2 bw1[9];
    #pragma unroll
    for (int s = 0; s < 9; ++s)
        #pragma unroll
        for (int r = 0; r < 2; ++r) {
            int k = 4 * s + khalf + r;            // padded K index 0..35
            int row = (k < 15) ? k : k - 1;       // undo pad column
            bw1[s][r] = (k != 15 && n < 10) ? W1[row * 10 + n] : 0.0f;
        }
    f2 bw2[4];
    #pragma unroll
    for (int s = 0; s < 4; ++s)
        #pragma unroll
        for (int r = 0; r < 2; ++r) {
            int k = 4 * s + khalf + r;
            bw2[s][r] = (k < 10 && n < 10) ? W2[k * 10 + n] : 0.0f;
        }
    const float bias1 = (n < 10) ? b1[n] : 0.0f;
    const float bias2 = (n < 10) ? b2[n] : 0.0f;

    for (int it = 0; it < iters; ++it) {
        const int tile = wave_id + it * total_waves;
        const int ebase = tile * TILE;

        // ---- stage concatenated feature tile [16 x 36] into wave-private LDS ----
        if (lane < 16) {
            int e = ebase + lane; if (e >= E) e = E - 1;          // clamped safe read
            int s = __builtin_nontemporal_load(&edge_index[e]);   // row 0 = src
            const f2* xs2 = (const f2*)(x_s + (long long)s * 10);
            f2* d0 = (f2*)(t1 + lane * LDS1_STRIDE);              // cols 0-9 (8B aligned)
            #pragma unroll
            for (int c = 0; c < 5; ++c) d0[c] = xs2[c];
            const f2* ea2 = (const f2*)(edge_attr + (long long)e * 10);
            f2* d1 = (f2*)(t1 + lane * LDS1_STRIDE + 16);         // cols 16-25 (8B aligned)
            #pragma unroll
            for (int c = 0; c < 5; ++c) d1[c] = __builtin_nontemporal_load(&ea2[c]);
        } else {
            int row = lane - 16;
            int e = ebase + row; if (e >= E) e = E - 1;
            int t = __builtin_nontemporal_load(&edge_index[E + e]); // row 1 = tgt
            const float* xt = x_t + (long long)t * 5;               // only 4B aligned
            float v0 = xt[0], v1 = xt[1], v2 = xt[2], v3 = xt[3], v4 = xt[4];
            f2* d0 = (f2*)(t1 + row * LDS1_STRIDE + 10);          // cols 10-13 (8B aligned)
            f2 p01 = {v0, v1}, p23 = {v2, v3}, p4z = {v4, 0.0f};
            d0[0] = p01; d0[1] = p23;
            ((f2*)(t1 + row * LDS1_STRIDE + 14))[0] = p4z;        // cols 14,15 (pad=0)
            int b = __builtin_nontemporal_load(&batch_e[e]);
            const f2* u2 = (const f2*)(u + (long long)b * 10);
            f2* d2 = (f2*)(t1 + row * LDS1_STRIDE + 26);          // cols 26-35 (8B aligned)
            #pragma unroll
            for (int c = 0; c < 5; ++c) d2[c] = u2[c];
        }

        // wave-private tile: in-order LDS per wave + explicit dscnt fence
        asm volatile("s_wait_dscnt 0x0" ::: "memory");

        // ---- layer 1: C[16x16] = A[16x36] x W1pad[36x16], 9 WMMA K-steps ----
        v8f acc = {};
        #pragma unroll
        for (int s = 0; s < 9; ++s) {
            int k = 4 * s + khalf;                                // even -> 8B aligned
            f2 a = *(const f2*)(t1 + m * LDS1_STRIDE + k);
            acc = __builtin_amdgcn_wmma_f32_16x16x4_f32(
                      false, a, false, bw1[s], (short)0, acc, false, false);
        }

        // ---- bias + leaky_relu(0.1); reshape C-layout -> row-major in LDS ----
        #pragma unroll
        for (int j = 0; j < 8; ++j) {
            int row = j + ((lane >> 4) ? 8 : 0);  // C layout: VGPR j -> M=j or j+8
            float v = acc[j] + bias1;
            v = (v > 0.0f) ? v : 0.1f * v;        // pad columns stay exactly 0
            t2[row * LDS2_STRIDE + n] = v;
        }

        asm volatile("s_wait_dscnt 0x0" ::: "memory");

        // ---- layer 2: C[16x16] = H1[16x16] x W2pad[16x16], 4 WMMA K-steps ----
        v8f acc2 = {};
        #pragma unroll
        for (int s = 0; s < 4; ++s) {
            int k = 4 * s + khalf;
            f2 a = *(const f2*)(t2 + m * LDS2_STRIDE + k);
            acc2 = __builtin_amdgcn_wmma_f32_16x16x4_f32(
                       false, a, false, bw2[s], (short)0, acc2, false, false);
        }

        // ---- store [16 x 10] non-temporally (cols 10-15 are padding) ----
        if (n < 10) {
            #pragma unroll
            for (int j = 0; j < 8; ++j) {
                int row = j + ((lane >> 4) ? 8 : 0);
                int e = ebase + row;
                if (e < E)
                    __builtin_nontemporal_store(acc2[j] + bias2,
                                                &out[(long long)e * 10 + n]);
            }
        }
    }
}

extern "C" void kernel_launch(void* const* d_in, const int* in_sizes, int n_in,
                              void* d_out, int out_size, void* d_ws, size_t ws_size,
                              hipStream_t stream) {
    const float* x_s       = (const float*)d_in[0];
    const float* x_t       = (const float*)d_in[1];
    const int*   edge_idx  = (const int*)  d_in[2];
    const float* edge_attr = (const float*)d_in[3];
    const float* u         = (const float*)d_in[4];
    const int*   batch_e   = (const int*)  d_in[5];
    const float* W1        = (const float*)d_in[6];
    const float* b1        = (const float*)d_in[7];
    const float* W2        = (const float*)d_in[8];
    const float* b2        = (const float*)d_in[9];
    float*       out       = (float*)d_out;

    const int E = in_sizes[5];                       // batch_e has E elements
    const int tiles = (E + TILE - 1) / TILE;         // 125000 for E=2M
    const int waves = (tiles + TILES_PER_WAVE - 1) / TILES_PER_WAVE;
    const int blocks = (waves + WAVES_PER_BLOCK - 1) / WAVES_PER_BLOCK;

    edge_mlp_wmma<<<blocks, WAVES_PER_BLOCK * 32, 0, stream>>>(
        x_s, x_t, edge_idx, edge_attr, u, batch_e, W1, b1, W2, b2, out, E, tiles);
}